// DataAugmentation_4827543240898
// MI455X (gfx1250) — compile-verified
//
#include <hip/hip_runtime.h>
#include <stdint.h>

// Fused data augmentation for MI455X (gfx1250).
// Roofline: 4096 images x (48KB in + 48KB out) = 403 MB @ 23.3 TB/s => ~17us
// floor; ~2 GFLOP VALU is negligible => pure bandwidth problem.
// Strategy: 1 workgroup per image; CDNA5 async global->LDS staging
// (global_load_async_to_lds_b128 / s_wait_asynccnt) so the 16K bilinear
// gathers per image hit LDS; per-image gray mean via wave32 shfl butterfly +
// ds_add_f32; each thread owns 4 quads of 4 consecutive pixels so outputs are
// written with global_store_b128.

#define OUTDIM 64
#define IMG_PIX (OUTDIM * OUTDIM)        // 4096
#define IMG_ELEMS (3 * IMG_PIX)          // 12288 floats = 48 KB
#define THREADS 256
#define QUADS 4                          // 4 quads * 4 px = 16 px/thread

__device__ __forceinline__ float clamp01(float v) {
    return fminf(fmaxf(v, 0.0f), 1.0f);
}

__global__ __launch_bounds__(THREADS)
void aug_kernel(const float* __restrict__ x,
                const int*   __restrict__ flip_mask,
                const float* __restrict__ ci_,
                const float* __restrict__ cj_,
                const float* __restrict__ ch_,
                const float* __restrict__ cw_,
                const float* __restrict__ bf_,
                const float* __restrict__ cf_,
                const float* __restrict__ sf_,
                const float* __restrict__ hf_,
                float* __restrict__ out)
{
    __shared__ float simg[IMG_ELEMS];   // staged source image (48 KB)
    __shared__ float ssum;              // gray-mean accumulator

    const int b = blockIdx.x;
    const int t = threadIdx.x;

    const float* src = x + (size_t)b * IMG_ELEMS;

    // ---- CDNA5 async copy: global -> LDS, 128b per lane, 12 x 4KB waves ----
    {
        uint32_t lds_base = (uint32_t)(uintptr_t)(&simg[0]);
        #pragma unroll
        for (int i = 0; i < (IMG_ELEMS / 4) / THREADS; ++i) {   // 12 iterations
            uint32_t byteoff = (uint32_t)(t + i * THREADS) * 16u;
            uint32_t ldsaddr = lds_base + byteoff;
            asm volatile("global_load_async_to_lds_b128 %0, %1, %2"
                         :: "v"(ldsaddr), "v"(byteoff), "s"(src)
                         : "memory");
        }
    }
    if (t == 0) ssum = 0.0f;
    asm volatile("s_wait_asynccnt 0" ::: "memory");
    __syncthreads();

    // ---- per-image uniforms ----
    const float ci = ci_[b], cj = cj_[b], ch = ch_[b], cw = cw_[b];
    const float bf = bf_[b], cf = cf_[b], sf = sf_[b], hf = hf_[b];
    const int   flip = flip_mask[b];

    const float sy = ch * (1.0f / OUTDIM);
    const float sx = cw * (1.0f / OUTDIM);

    float rr[QUADS][4], gg[QUADS][4], bb[QUADS][4];
    float psum = 0.0f;

    // ---- resized crop (bilinear from LDS) + flip + brightness + gray sum ----
    #pragma unroll
    for (int q = 0; q < QUADS; ++q) {
        const int p0  = q * (THREADS * 4) + t * 4;   // quad base pixel
        const int row = p0 >> 6;                     // same row for all 4 px
        const int col0 = p0 & 63;

        // y path shared by the whole quad
        const float ys = ci + ((float)row + 0.5f) * sy - 0.5f;
        const float y0 = floorf(ys);
        const float wy = ys - y0;
        const int y0i = min(max((int)y0, 0), 63);
        const int y1i = min(y0i + 1, 63);

        #pragma unroll
        for (int k = 0; k < 4; ++k) {
            const int col = col0 + k;
            const float xs = cj + ((float)col + 0.5f) * sx - 0.5f;
            const float x0 = floorf(xs);
            const float wx = xs - x0;
            int x0i = min(max((int)x0, 0), 63);
            int x1i = min(x0i + 1, 63);
            if (flip) { x0i = 63 - x0i; x1i = 63 - x1i; }  // sample flipped img

            float v3[3];
            #pragma unroll
            for (int c = 0; c < 3; ++c) {
                const float* pc = &simg[c * IMG_PIX];
                const float a  = pc[y0i * 64 + x0i];
                const float bq = pc[y0i * 64 + x1i];
                const float cc = pc[y1i * 64 + x0i];
                const float dd = pc[y1i * 64 + x1i];
                const float top = a  * (1.0f - wx) + bq * wx;
                const float bot = cc * (1.0f - wx) + dd * wx;
                float val = top * (1.0f - wy) + bot * wy;
                v3[c] = clamp01(val * bf);               // brightness
            }
            rr[q][k] = v3[0]; gg[q][k] = v3[1]; bb[q][k] = v3[2];
            psum += 0.2989f * v3[0] + 0.587f * v3[1] + 0.114f * v3[2];
        }
    }

    // ---- per-image mean: wave32 butterfly reduce + LDS float atomic ----
    #pragma unroll
    for (int off = 16; off > 0; off >>= 1)
        psum += __shfl_xor(psum, off, 32);
    if ((t & 31) == 0) atomicAdd(&ssum, psum);           // ds_add_f32
    __syncthreads();
    const float mean = ssum * (1.0f / (float)IMG_PIX);

    // hoisted uniforms
    const float cmix = (1.0f - cf) * mean;
    const float sres = (1.0f - sf);

    // ---- contrast, saturation, hue, clamp; vectorized b128 stores ----
    float* ob = out + (size_t)b * IMG_ELEMS;
    #pragma unroll
    for (int q = 0; q < QUADS; ++q) {
        float4 Rv, Gv, Bv;
        float* Rp = &Rv.x; float* Gp = &Gv.x; float* Bp = &Bv.x;

        #pragma unroll
        for (int k = 0; k < 4; ++k) {
            float r = rr[q][k], g = gg[q][k], bl = bb[q][k];

            // contrast
            r  = clamp01(cf * r  + cmix);
            g  = clamp01(cf * g  + cmix);
            bl = clamp01(cf * bl + cmix);

            // saturation
            const float gray = 0.2989f * r + 0.587f * g + 0.114f * bl;
            r  = clamp01(sf * r  + sres * gray);
            g  = clamp01(sf * g  + sres * gray);
            bl = clamp01(sf * bl + sres * gray);

            // hue rotation (RGB -> H, shift, -> RGB), matches reference math
            const float maxc = fmaxf(r, fmaxf(g, bl));
            const float minc = fminf(r, fminf(g, bl));
            const float hv   = maxc;
            const float cr   = maxc - minc;
            const float hs   = cr / ((maxc == 0.0f) ? 1.0f : maxc);
            const float crd  = (cr == 0.0f) ? 1.0f : cr;
            const float rc   = (maxc - r)  / crd;
            const float gc   = (maxc - g)  / crd;
            const float bc   = (maxc - bl) / crd;
            float h = (maxc == r) ? (bc - gc)
                     : ((maxc == g) ? (2.0f + rc - bc) : (4.0f + gc - rc));
            h = h * (1.0f / 6.0f);
            h = (cr == 0.0f) ? 0.0f : (h - floorf(h));   // mod 1
            h = h + hf;
            h = h - floorf(h);                            // mod 1

            const float h6 = h * 6.0f;
            const float i6 = floorf(h6);
            const float f  = h6 - i6;
            const float hp = hv * (1.0f - hs);
            const float hq = hv * (1.0f - f * hs);
            const float ht = hv * (1.0f - (1.0f - f) * hs);
            const int ii = ((int)i6) % 6;

            float ro = (ii == 0) ? hv : (ii == 1) ? hq : (ii == 2) ? hp
                     : (ii == 3) ? hp : (ii == 4) ? ht : hv;
            float go = (ii == 0) ? ht : (ii == 1) ? hv : (ii == 2) ? hv
                     : (ii == 3) ? hq : (ii == 4) ? hp : hp;
            float bo = (ii == 0) ? hp : (ii == 1) ? hp : (ii == 2) ? ht
                     : (ii == 3) ? hv : (ii == 4) ? hv : hq;

            Rp[k] = clamp01(ro);
            Gp[k] = clamp01(go);
            Bp[k] = clamp01(bo);
        }

        const int p0 = q * (THREADS * 4) + t * 4;        // 16B aligned
        *(float4*)(ob + 0 * IMG_PIX + p0) = Rv;          // global_store_b128
        *(float4*)(ob + 1 * IMG_PIX + p0) = Gv;
        *(float4*)(ob + 2 * IMG_PIX + p0) = Bv;
    }
}

extern "C" void kernel_launch(void* const* d_in, const int* in_sizes, int n_in,
                              void* d_out, int out_size, void* d_ws, size_t ws_size,
                              hipStream_t stream) {
    const float* x    = (const float*)d_in[0];
    const int*   flip = (const int*)  d_in[1];
    const float* ci   = (const float*)d_in[2];
    const float* cj   = (const float*)d_in[3];
    const float* ch   = (const float*)d_in[4];
    const float* cw   = (const float*)d_in[5];
    const float* bf   = (const float*)d_in[6];
    const float* cf   = (const float*)d_in[7];
    const float* sf   = (const float*)d_in[8];
    const float* hf   = (const float*)d_in[9];
    float* out = (float*)d_out;

    const int B = in_sizes[2];   // 4096 (crop_i length)
    aug_kernel<<<B, THREADS, 0, stream>>>(x, flip, ci, cj, ch, cw, bf, cf, sf, hf, out);
}